// AdaptiveGN_Patches_Hadamart_75144747811287
// MI455X (gfx1250) — compile-verified
//
#include <hip/hip_runtime.h>
#include <hip/hip_bf16.h>
#include <math.h>

// ---------------------------------------------------------------------------
// AdaptiveGN_Patches_Hadamard for MI455X (gfx1250)
//   x,y: (8,128,256,256) f32.  Per (b, patch_i, patch_j, group) GroupNorm over
//   4ch x 64x64 = 16384 floats, then out = xn * (1 + silu(y)).
//   Memory-bound (805 MB min traffic -> ~35us @ 23.3 TB/s): stage the 64KB
//   region in LDS via the Tensor Data Mover so x is read from HBM exactly
//   once; prefetch y into L2 while the DMA is in flight; NT stores for out.
// ---------------------------------------------------------------------------

#define BLOCK 256
constexpr int P       = 64;     // patch edge
constexpr int CG      = 4;      // channels per group (128/32)
constexpr int HW      = 256;    // image edge
constexpr int CHANS   = 128;
constexpr int TILE_EL = CG * P * P;          // 16384 floats = 64 KB
constexpr float EPS   = 1e-5f;
constexpr float INV_N = 1.0f / (float)TILE_EL;

typedef unsigned int v4u __attribute__((ext_vector_type(4)));
typedef int          v4i __attribute__((ext_vector_type(4)));
typedef int          v8i __attribute__((ext_vector_type(8)));
typedef float        v4f __attribute__((ext_vector_type(4)));   // native vec4

#if defined(__has_builtin)
#if __has_builtin(__builtin_amdgcn_tensor_load_to_lds)
#define HAVE_TDM 1
#endif
#if __has_builtin(__builtin_amdgcn_exp2f) && __has_builtin(__builtin_amdgcn_rcpf)
#define HAVE_FASTMATH 1
#endif
#endif
#ifndef HAVE_TDM
#define HAVE_TDM 0
#endif
#ifndef HAVE_FASTMATH
#define HAVE_FASTMATH 0
#endif

// 1 + silu(v) with raw v_exp_f32 / v_rcp_f32 (~7 VALU ops incl. the fma)
__device__ __forceinline__ float gate1p(float v) {
#if HAVE_FASTMATH
    const float t = __builtin_amdgcn_exp2f(-1.442695040888963f * v); // exp(-v)
    return fmaf(v, __builtin_amdgcn_rcpf(1.0f + t), 1.0f);
#else
    return 1.0f + v / (1.0f + __expf(-v));
#endif
}

__global__ __launch_bounds__(BLOCK)
void agn_patches_kernel(const float* __restrict__ x,
                        const float* __restrict__ y,
                        const float* __restrict__ weight,
                        const float* __restrict__ bias,
                        float* __restrict__ out)
{
    __shared__ float tile[TILE_EL];      // x region staged by TDM
    __shared__ float red1[8], red2[8];
    __shared__ float s_mean, s_rstd;

    // region decode: blockIdx.x = ((b*4 + i)*4 + j)*32 + grp
    unsigned r   = blockIdx.x;
    const unsigned grp = r & 31u;  r >>= 5;
    const unsigned j   = r & 3u;   r >>= 2;
    const unsigned i   = r & 3u;   r >>= 2;
    const unsigned b   = r;
    const unsigned c0  = grp * CG;
    // flat float offset of (b, c0, i*64, j*64) in x
    const unsigned long long base =
        (((unsigned long long)b * CHANS + c0) * HW + (unsigned long long)i * P) * HW
        + (unsigned long long)j * P;

    const unsigned tid  = threadIdx.x;
    const unsigned wave = tid >> 5;

#if HAVE_TDM
    // --- Tensor Data Mover: waves 0..3 each DMA one channel's 64x64 tile ---
    if (wave < CG) {
        const unsigned long long ga =
            (unsigned long long)(const void*)(x + base + (unsigned long long)wave * (HW * HW));
        const unsigned lds_off =
            (unsigned)(unsigned long long)(&tile[wave * (P * P)]);
        // D# group 0: count=1 | lds_addr | global_addr | type=2
        v4u g0 = { 1u,
                   lds_off,
                   (unsigned)ga,
                   (unsigned)((ga >> 32) & 0x1FFFFFFull) | (2u << 30) };
        // D# group 1: data_size=4B; tensor_dim0=64, dim1=64; tile 64x64x1;
        //             dim0_stride=256, dim1_stride=65536
        v8i g1 = { (int)(2u << 16),            // data_size=2 (4B)
                   (int)((unsigned)P << 16),   // tensor_dim0 lo16
                   (int)((unsigned)P << 16),   // dim0 hi=0 | tensor_dim1 lo16
                   (int)((unsigned)P << 16),   // dim1 hi=0 | tile_dim0=64
                   (int)(P | (1u << 16)),      // tile_dim1=64 | tile_dim2=1
                   (int)HW,                    // tensor_dim0_stride = 256
                   0,                          // strides mid bits = 0
                   (int)((HW * HW) >> 16) };   // tensor_dim1_stride[47:16]
        v4i g2 = { 1, 1, 0, 0 };               // tensor_dim2=1, tensor_dim3=1
        v4i g3 = { 0, 0, 0, 0 };
        v8i gpad = { 0, 0, 0, 0, 0, 0, 0, 0 }; // 5th arg on clang-23 lane
        __builtin_amdgcn_tensor_load_to_lds(g0, g1, g2, g3, gpad, 0);
    }
    // --- all waves: warm L2 with this block's y region while the DMA runs ---
    #pragma unroll
    for (int k = 0; k < TILE_EL / (BLOCK * 4); ++k) {
        const unsigned f  = (unsigned)k * BLOCK + tid;
        const unsigned e  = f * 4;
        const unsigned cc = (unsigned)k >> 2;
        const unsigned hp = (e >> 6) & (P - 1);
        const unsigned wx = e & (P - 1);
        const unsigned long long ga = base
            + (unsigned long long)cc * (HW * HW) + (unsigned long long)hp * HW + wx;
        __builtin_prefetch(&y[ga], 0, 1);      // global_prefetch_b8
    }
    if (wave < CG) {
        __builtin_amdgcn_s_wait_tensorcnt(0);
    }
#else
    // Fallback: cooperative load of the region into LDS (still 1 HBM pass).
    #pragma unroll
    for (int k = 0; k < TILE_EL / (BLOCK * 4); ++k) {
        const unsigned f  = (unsigned)k * BLOCK + tid;   // float4 index
        const unsigned e  = f * 4;
        const unsigned cc = e >> 12;
        const unsigned hp = (e >> 6) & (P - 1);
        const unsigned wx = e & (P - 1);
        const unsigned long long ga = base
            + (unsigned long long)cc * (HW * HW) + (unsigned long long)hp * HW + wx;
        *reinterpret_cast<v4f*>(&tile[e]) =
            *reinterpret_cast<const v4f*>(&x[ga]);
    }
#endif
    __syncthreads();

    // --- pass 1: sum / sum-of-squares over the 16384-float region ---
    float s1 = 0.f, s2 = 0.f;
    #pragma unroll
    for (int k = 0; k < TILE_EL / (BLOCK * 4); ++k) {
        const v4f v = *reinterpret_cast<const v4f*>(&tile[((unsigned)k * BLOCK + tid) * 4]);
        s1 += (v.x + v.y) + (v.z + v.w);
        s2 += (v.x * v.x + v.y * v.y) + (v.z * v.z + v.w * v.w);
    }
    #pragma unroll
    for (int off = 16; off > 0; off >>= 1) {
        s1 += __shfl_xor(s1, off, 32);
        s2 += __shfl_xor(s2, off, 32);
    }
    if ((tid & 31u) == 0u) { red1[wave] = s1; red2[wave] = s2; }
    __syncthreads();
    if (tid == 0) {
        float t1 = 0.f, t2 = 0.f;
        #pragma unroll
        for (int q = 0; q < 8; ++q) { t1 += red1[q]; t2 += red2[q]; }
        const float mean = t1 * INV_N;
        const float var  = fmaf(-mean, mean, t2 * INV_N);
        s_mean = mean;
        s_rstd = rsqrtf(var + EPS);
    }
    __syncthreads();
    const float mean = s_mean;
    const float rstd = s_rstd;

    // --- pass 2: normalize from LDS, gate with silu(y), stream out (NT) ---
    #pragma unroll
    for (int k = 0; k < TILE_EL / (BLOCK * 4); ++k) {
        const unsigned f  = (unsigned)k * BLOCK + tid;
        const unsigned e  = f * 4;
        const unsigned cc = (unsigned)k >> 2;             // channel-in-group
        const unsigned hp = (e >> 6) & (P - 1);
        const unsigned wx = e & (P - 1);
        const unsigned long long ga = base
            + (unsigned long long)cc * (HW * HW) + (unsigned long long)hp * HW + wx;

        const float wch = weight[c0 + cc];
        const float sw  = wch * rstd;                      // scale
        const float sb  = fmaf(-mean, sw, bias[c0 + cc]);  // shift

        const v4f xv = *reinterpret_cast<const v4f*>(&tile[e]);
        const v4f yv = *reinterpret_cast<const v4f*>(&y[ga]);
        v4f o;
        o.x = fmaf(xv.x, sw, sb) * gate1p(yv.x);
        o.y = fmaf(xv.y, sw, sb) * gate1p(yv.y);
        o.z = fmaf(xv.z, sw, sb) * gate1p(yv.z);
        o.w = fmaf(xv.w, sw, sb) * gate1p(yv.w);
        __builtin_nontemporal_store(o, reinterpret_cast<v4f*>(&out[ga]));
    }
}

extern "C" void kernel_launch(void* const* d_in, const int* in_sizes, int n_in,
                              void* d_out, int out_size, void* d_ws, size_t ws_size,
                              hipStream_t stream) {
    (void)in_sizes; (void)n_in; (void)out_size; (void)d_ws; (void)ws_size;
    const float* x      = (const float*)d_in[0];
    const float* y      = (const float*)d_in[1];
    const float* weight = (const float*)d_in[2];
    const float* bias   = (const float*)d_in[3];
    float* out          = (float*)d_out;

    const int n_regions = 8 * 4 * 4 * 32;   // b * i * j * groups = 4096
    agn_patches_kernel<<<n_regions, BLOCK, 0, stream>>>(x, y, weight, bias, out);
}